// EnergyModel_45251775430723
// MI455X (gfx1250) — compile-verified
//
#include <hip/hip_runtime.h>
#include <hip/hip_bf16.h>
#include <math.h>

// ---------------- problem constants (match reference setup_inputs) ----------
#define NATOM 4096
#define N0A   2048
#define NNEI_ 138
#define SEL0  46
#define SEL1  92
#define M1_   100
#define M2_   16
#define FITW  240
#define DWID  1600          // M1*M2
#define NPAIR ((long)NATOM * NNEI_)

// output layout (floats): E | force(3*4096) | virial(9) | ae(4096) | atom_virial(9*4096)
#define E_OFF  0
#define F_OFF  1
#define V_OFF  (1 + 3*NATOM)
#define AE_OFF (V_OFF + 9)
#define AV_OFF (AE_OFF + NATOM)
#define OUT_TOTAL (AV_OFF + 9*NATOM)

typedef __attribute__((ext_vector_type(2))) float v2f;
typedef __attribute__((ext_vector_type(8))) float v8f;

// ---------------- helpers ---------------------------------------------------
__device__ __forceinline__ long rowmap(long r, int segLen, int segStride) {
  return segLen > 0 ? (r / segLen) * (long)segStride + (r % segLen) : r;
}

__device__ __forceinline__ float swfun(float r) {
  if (r < 0.5f) return 1.0f;
  if (r >= 6.0f) return 0.0f;
  float u = (r - 0.5f) / 5.5f;
  return u*u*u*(-6.0f*u*u + 15.0f*u - 10.0f) + 1.0f;
}
__device__ __forceinline__ float dswfun(float r) {
  if (r <= 0.5f || r >= 6.0f) return 0.0f;
  float u = (r - 0.5f) / 5.5f;
  return (u*u*(-30.0f*u*u + 60.0f*u - 30.0f)) / 5.5f;
}

// ---------------- WMMA f32 32x16 macro-tile (two 16x16 accumulators) --------
// Requirements: M % 32 == 0 (row guards dropped). Column edge + K tail handled
// with clamped addresses + selects (branch-free, loads always issued).
template <bool BT>
__device__ __forceinline__ void wmma_f32_tile2(const float* __restrict__ A, int lda,
                                               const float* __restrict__ B, int ldb,
                                               int K, long rowPhys0, long rowPhys1,
                                               int bcl, bool bcok,
                                               v8f& acc0, v8f& acc1) {
  const int half = (threadIdx.x >> 4) & 1;
  const float* __restrict__ pA0 = A + rowPhys0 * (long)lda;
  const float* __restrict__ pA1 = A + rowPhys1 * (long)lda;
  const int K4 = K & ~3;
  #pragma unroll 4
  for (int k0 = 0; k0 < K4; k0 += 4) {
    const int ak = k0 + half * 2;
    v2f a0, a1, b;
    a0.x = pA0[ak];     a0.y = pA0[ak + 1];
    a1.x = pA1[ak];     a1.y = pA1[ak + 1];
    float bx, by;
    if (!BT) { bx = B[(long)ak * ldb + bcl]; by = B[(long)(ak + 1) * ldb + bcl]; }
    else     { bx = B[(long)bcl * ldb + ak]; by = B[(long)bcl * ldb + ak + 1]; }
    b.x = bcok ? bx : 0.0f;
    b.y = bcok ? by : 0.0f;
    acc0 = __builtin_amdgcn_wmma_f32_16x16x4_f32(false, a0, false, b, (short)0, acc0, false, false);
    acc1 = __builtin_amdgcn_wmma_f32_16x16x4_f32(false, a1, false, b, (short)0, acc1, false, false);
  }
  if (K4 < K) {                      // K tail (e.g. K == 25)
    const int ak = K4 + half * 2;
    const int i0 = ak     < K ? ak     : K - 1;
    const int i1 = ak + 1 < K ? ak + 1 : K - 1;
    const bool ok0 = ak < K, ok1 = ak + 1 < K;
    v2f a0, a1, b;
    a0.x = ok0 ? pA0[i0] : 0.0f;  a0.y = ok1 ? pA0[i1] : 0.0f;
    a1.x = ok0 ? pA1[i0] : 0.0f;  a1.y = ok1 ? pA1[i1] : 0.0f;
    float bx, by;
    if (!BT) { bx = B[(long)i0 * ldb + bcl]; by = B[(long)i1 * ldb + bcl]; }
    else     { bx = B[(long)bcl * ldb + i0]; by = B[(long)bcl * ldb + i1]; }
    b.x = (bcok && ok0) ? bx : 0.0f;
    b.y = (bcok && ok1) ? by : 0.0f;
    acc0 = __builtin_amdgcn_wmma_f32_16x16x4_f32(false, a0, false, b, (short)0, acc0, false, false);
    acc1 = __builtin_amdgcn_wmma_f32_16x16x4_f32(false, a1, false, b, (short)0, acc1, false, false);
  }
}

// ---------------- GEMM kernels (one wave = one 32x16 tile) ------------------
// C = tanh(A@B + bias)
__global__ void k_gemm_tanh(const float* __restrict__ A, int lda,
                            const float* __restrict__ B, int ldb,
                            const float* __restrict__ bias,
                            float* __restrict__ C, int ldc,
                            int N, int K, int segLen, int segStride) {
  const int rowBase = blockIdx.y * 32, colBase = blockIdx.x * 16;
  const int lane = threadIdx.x & 31, half = lane >> 4, l16 = lane & 15;
  const int bc = colBase + l16;
  const bool bcok = bc < N;
  const int bcl = bcok ? bc : N - 1;
  const long rp0 = rowmap(rowBase + l16,      segLen, segStride);
  const long rp1 = rowmap(rowBase + 16 + l16, segLen, segStride);
  v8f acc0 = {}, acc1 = {};
  wmma_f32_tile2<false>(A, lda, B, ldb, K, rp0, rp1, bcl, bcok, acc0, acc1);
  if (!bcok) return;
  const float bv = bias[bc];
  for (int v = 0; v < 8; ++v) {
    int r0 = rowBase + v + half * 8;
    long p0 = rowmap(r0,      segLen, segStride);
    long p1 = rowmap(r0 + 16, segLen, segStride);
    C[p0 * ldc + bc] = tanhf(acc0[v] + bv);
    C[p1 * ldc + bc] = tanhf(acc1[v] + bv);
  }
}

// t = tanh(A@B + bias); H = t + A[:, col % K]  (residual concat, N == 2*K)
__global__ void k_gemm_tanh_resconcat(const float* __restrict__ A, int lda,
                                      const float* __restrict__ B, int ldb,
                                      const float* __restrict__ bias,
                                      float* __restrict__ Tb,
                                      float* __restrict__ H, int ldc,
                                      int N, int K, int segLen, int segStride) {
  const int rowBase = blockIdx.y * 32, colBase = blockIdx.x * 16;
  const int lane = threadIdx.x & 31, half = lane >> 4, l16 = lane & 15;
  const int bc = colBase + l16;
  const bool bcok = bc < N;
  const int bcl = bcok ? bc : N - 1;
  const long rp0 = rowmap(rowBase + l16,      segLen, segStride);
  const long rp1 = rowmap(rowBase + 16 + l16, segLen, segStride);
  v8f acc0 = {}, acc1 = {};
  wmma_f32_tile2<false>(A, lda, B, ldb, K, rp0, rp1, bcl, bcok, acc0, acc1);
  if (!bcok) return;
  const float bv = bias[bc];
  const int rcol = bc % K;
  for (int v = 0; v < 8; ++v) {
    int r0 = rowBase + v + half * 8;
    long p0 = rowmap(r0,      segLen, segStride);
    long p1 = rowmap(r0 + 16, segLen, segStride);
    float t0 = tanhf(acc0[v] + bv);
    float t1 = tanhf(acc1[v] + bv);
    Tb[p0 * ldc + bc] = t0;
    Tb[p1 * ldc + bc] = t1;
    H [p0 * ldc + bc] = t0 + A[p0 * (long)lda + rcol];
    H [p1 * ldc + bc] = t1 + A[p1 * (long)lda + rcol];
  }
}

// t = tanh(A@B + bias); H = A + idt*t   (fit residual, N == K == lda == ldc)
__global__ void k_gemm_fit_res(const float* __restrict__ A, int lda,
                               const float* __restrict__ B, int ldb,
                               const float* __restrict__ bias,
                               const float* __restrict__ idt,
                               float* __restrict__ Tb,
                               float* __restrict__ H, int ldc,
                               int N, int K) {
  const int rowBase = blockIdx.y * 32, colBase = blockIdx.x * 16;
  const int lane = threadIdx.x & 31, half = lane >> 4, l16 = lane & 15;
  const int bc = colBase + l16;
  const bool bcok = bc < N;
  const int bcl = bcok ? bc : N - 1;
  v8f acc0 = {}, acc1 = {};
  wmma_f32_tile2<false>(A, lda, B, ldb, K, rowBase + l16, rowBase + 16 + l16,
                        bcl, bcok, acc0, acc1);
  if (!bcok) return;
  const float bv = bias[bc], iv = idt[bc];
  for (int v = 0; v < 8; ++v) {
    long p0 = rowBase + v + half * 8;
    long p1 = p0 + 16;
    float t0 = tanhf(acc0[v] + bv);
    float t1 = tanhf(acc1[v] + bv);
    Tb[p0 * ldc + bc] = t0;
    Tb[p1 * ldc + bc] = t1;
    H [p0 * ldc + bc] = A[p0 * (long)lda + bc] + iv * t0;
    H [p1 * ldc + bc] = A[p1 * (long)lda + bc] + iv * t1;
  }
}

// C = (Cin ? Cin : 0) + A @ B^T   (backward: dX = dZ @ W^T); Cin may alias C
__global__ void k_gemm_bt(const float* __restrict__ A, int lda,
                          const float* __restrict__ B, int ldb,
                          const float* __restrict__ Cin,
                          float* __restrict__ C, int ldc,
                          int N, int K, int segLen, int segStride) {
  const int rowBase = blockIdx.y * 32, colBase = blockIdx.x * 16;
  const int lane = threadIdx.x & 31, half = lane >> 4, l16 = lane & 15;
  const int bc = colBase + l16;
  const bool bcok = bc < N;
  const int bcl = bcok ? bc : N - 1;
  const long rp0 = rowmap(rowBase + l16,      segLen, segStride);
  const long rp1 = rowmap(rowBase + 16 + l16, segLen, segStride);
  v8f acc0 = {}, acc1 = {};
  wmma_f32_tile2<true>(A, lda, B, ldb, K, rp0, rp1, bcl, bcok, acc0, acc1);
  if (!bcok) return;
  for (int v = 0; v < 8; ++v) {
    int r0 = rowBase + v + half * 8;
    long p0 = rowmap(r0,      segLen, segStride);
    long p1 = rowmap(r0 + 16, segLen, segStride);
    float b0 = Cin ? Cin[p0 * ldc + bc] : 0.0f;
    float b1 = Cin ? Cin[p1 * ldc + bc] : 0.0f;
    C[p0 * ldc + bc] = b0 + acc0[v];
    C[p1 * ldc + bc] = b1 + acc1[v];
  }
}

// ---------------- non-GEMM kernels ------------------------------------------
__global__ void k_zero(float* p, long n) {
  long i = (long)blockIdx.x * blockDim.x + threadIdx.x;
  if (i < n) p[i] = 0.0f;
}

// one block per atom: candidate gather + rank selection in LDS
__global__ void k_build_nbr(const float* __restrict__ coord, int* __restrict__ nbr) {
  __shared__ float cd2[256];
  __shared__ int   cidx[256];
  __shared__ int   cnt;
  int n = blockIdx.x;
  float cx = coord[n*3+0], cy = coord[n*3+1], cz = coord[n*3+2];
  for (int j = threadIdx.x; j < NNEI_; j += blockDim.x) nbr[(long)n*NNEI_ + j] = -1;
  int base = 0;
  for (int rng = 0; rng < 2; ++rng) {
    int off   = rng ? N0A : 0;
    int count = rng ? (NATOM - N0A) : N0A;
    int kSel  = rng ? SEL1 : SEL0;
    if (threadIdx.x == 0) cnt = 0;
    __syncthreads();
    for (int j = off + threadIdx.x; j < off + count; j += blockDim.x) {
      if (j == n) continue;
      float dx = coord[j*3+0]-cx, dy = coord[j*3+1]-cy, dz = coord[j*3+2]-cz;
      float d2 = dx*dx + dy*dy + dz*dz;
      if (d2 < 36.0f) {
        int pos = atomicAdd(&cnt, 1);
        if (pos < 256) { cd2[pos] = d2; cidx[pos] = j; }
      }
    }
    __syncthreads();
    int c = cnt < 256 ? cnt : 256;
    for (int t = threadIdx.x; t < c; t += blockDim.x) {
      float dt_ = cd2[t]; int it = cidx[t];
      int rank = 0;
      for (int u = 0; u < c; ++u)
        if (cd2[u] < dt_ || (cd2[u] == dt_ && cidx[u] < it)) ++rank;
      if (rank < kSel) nbr[(long)n*NNEI_ + base + rank] = it;
    }
    __syncthreads();
    base += kSel;
  }
}

// per (atom, neighbor slot): rij, standardized descriptor Rt, ss
__global__ void k_build_R(const float* __restrict__ coord, const int* __restrict__ nbr,
                          const float* __restrict__ davg, const float* __restrict__ dstd,
                          float* __restrict__ rij, float* __restrict__ Rt,
                          float* __restrict__ ss) {
  long i = (long)blockIdx.x * blockDim.x + threadIdx.x;
  if (i >= NPAIR) return;
  int n = (int)(i / NNEI_), k = (int)(i % NNEI_);
  int ti = n < N0A ? 0 : 1;
  int idx = nbr[i];
  float R[4] = {0,0,0,0};
  float rx = 0, ry = 0, rz = 0;
  if (idx >= 0) {
    rx = coord[idx*3+0]-coord[n*3+0];
    ry = coord[idx*3+1]-coord[n*3+1];
    rz = coord[idx*3+2]-coord[n*3+2];
    float r2 = rx*rx + ry*ry + rz*rz;
    float r  = sqrtf(r2);
    float s  = swfun(r) / r;
    R[0] = s; R[1] = s/r*rx; R[2] = s/r*ry; R[3] = s/r*rz;
  }
  rij[i*3+0] = rx; rij[i*3+1] = ry; rij[i*3+2] = rz;
  for (int a = 0; a < 4; ++a) {
    long di = (long)ti * NNEI_ * 4 + k*4 + a;
    float v = (R[a] - davg[di]) / dstd[di];
    Rt[i*4 + a] = v;
    if (a == 0) ss[i] = v;
  }
}

__global__ void k_emb_l1(const float* __restrict__ ss, const float* __restrict__ w1,
                         const float* __restrict__ b1, float* __restrict__ h1) {
  long i = (long)blockIdx.x * blockDim.x + threadIdx.x;
  if (i >= NPAIR) return;
  int n = (int)(i / NNEI_), k = (int)(i % NNEI_);
  int p = (n < N0A ? 0 : 2) + (k >= SEL0 ? 1 : 0);
  float s = ss[i];
  for (int j = 0; j < 25; ++j)
    h1[i*25 + j] = tanhf(s * w1[p*25 + j] + b1[p*25 + j]);
}

__global__ void k_T(const float* __restrict__ Rt, const float* __restrict__ G,
                    float* __restrict__ T) {
  long t = (long)blockIdx.x * blockDim.x + threadIdx.x;
  if (t >= (long)NATOM * 400) return;
  int n = (int)(t / 400), a = (int)((t % 400) / 100), m = (int)(t % 100);
  float acc = 0.0f;
  long base = (long)n * NNEI_;
  for (int k = 0; k < NNEI_; ++k)
    acc += Rt[(base + k)*4 + a] * G[(base + k)*100 + m];
  T[t] = acc * (1.0f / NNEI_);
}

__global__ void k_D(const float* __restrict__ T, float* __restrict__ D) {
  long t = (long)blockIdx.x * blockDim.x + threadIdx.x;
  if (t >= (long)NATOM * DWID) return;
  int n = (int)(t / DWID), m = (int)((t % DWID) / M2_), b = (int)(t % M2_);
  float acc = 0.0f;
  for (int a = 0; a < 4; ++a)
    acc += T[(long)n*400 + a*100 + m] * T[(long)n*400 + a*100 + b];
  D[t] = acc;
}

__global__ void k_ae(const float* __restrict__ h2f, const float* __restrict__ wf,
                     const float* __restrict__ bf, const float* __restrict__ bae,
                     float* __restrict__ out) {
  int n = blockIdx.x * blockDim.x + threadIdx.x;
  if (n >= NATOM) return;
  int ti = n < N0A ? 0 : 1;
  float acc = bf[ti] + bae[ti];
  for (int j = 0; j < FITW; ++j)
    acc += h2f[(long)n*FITW + j] * wf[ti*FITW + j];
  out[AE_OFF + n] = acc;
  atomicAdd(&out[E_OFF], acc);
}

__global__ void k_dh2f_init(const float* __restrict__ wf, float* __restrict__ dh2f) {
  long i = (long)blockIdx.x * blockDim.x + threadIdx.x;
  if (i >= (long)NATOM * FITW) return;
  int n = (int)(i / FITW), j = (int)(i % FITW);
  int ti = n < N0A ? 0 : 1;
  dh2f[i] = wf[ti*FITW + j];
}

// u = g * (1-t^2) * (idt? idt[ti][j] : 1)
__global__ void k_fit_tanhgrad(const float* __restrict__ g, const float* __restrict__ t,
                               const float* __restrict__ idt, float* __restrict__ u) {
  long i = (long)blockIdx.x * blockDim.x + threadIdx.x;
  if (i >= (long)NATOM * FITW) return;
  int n = (int)(i / FITW), j = (int)(i % FITW);
  int ti = n < N0A ? 0 : 1;
  float tv = t[i];
  float v = g[i] * (1.0f - tv*tv);
  if (idt) v *= idt[ti*FITW + j];
  u[i] = v;
}

__global__ void k_emb_tanhgrad(const float* __restrict__ g, const float* __restrict__ t,
                               float* __restrict__ u, long n) {
  long i = (long)blockIdx.x * blockDim.x + threadIdx.x;
  if (i >= n) return;
  float tv = t[i];
  u[i] = g[i] * (1.0f - tv*tv);
}

// out[row, c] = in[row, c] + in[row, c+W]   (backward of concat([h,h]))
__global__ void k_reduce_concat(const float* __restrict__ in, float* __restrict__ out,
                                int W, long rows) {
  long i = (long)blockIdx.x * blockDim.x + threadIdx.x;
  if (i >= rows * W) return;
  long row = i / W; int c = (int)(i % W);
  out[i] = in[row*2*W + c] + in[row*2*W + c + W];
}

__global__ void k_dT_from_dD(const float* __restrict__ dD, const float* __restrict__ T,
                             float* __restrict__ dT) {
  long t = (long)blockIdx.x * blockDim.x + threadIdx.x;
  if (t >= (long)NATOM * 400) return;
  int n = (int)(t / 400), a = (int)((t % 400) / 100), m = (int)(t % 100);
  float v = 0.0f;
  for (int b = 0; b < M2_; ++b)
    v += dD[(long)n*DWID + m*M2_ + b] * T[(long)n*400 + a*100 + b];
  if (m < M2_)
    for (int mm = 0; mm < 100; ++mm)
      v += dD[(long)n*DWID + mm*M2_ + m] * T[(long)n*400 + a*100 + mm];
  dT[t] = v;
}

__global__ void k_dG(const float* __restrict__ Rt, const float* __restrict__ dT,
                     float* __restrict__ dG) {
  long t = (long)blockIdx.x * blockDim.x + threadIdx.x;
  if (t >= NPAIR * 100) return;
  long row = t / 100; int m = (int)(t % 100);
  int n = (int)(row / NNEI_);
  float v = 0.0f;
  for (int a = 0; a < 4; ++a)
    v += Rt[row*4 + a] * dT[(long)n*400 + a*100 + m];
  dG[t] = v * (1.0f / NNEI_);
}

__global__ void k_emb_l1_bwd(const float* __restrict__ dh1, const float* __restrict__ h1,
                             const float* __restrict__ w1, float* __restrict__ dss) {
  long i = (long)blockIdx.x * blockDim.x + threadIdx.x;
  if (i >= NPAIR) return;
  int n = (int)(i / NNEI_), k = (int)(i % NNEI_);
  int p = (n < N0A ? 0 : 2) + (k >= SEL0 ? 1 : 0);
  float acc = 0.0f;
  for (int j = 0; j < 25; ++j) {
    float h = h1[i*25 + j];
    acc += dh1[i*25 + j] * (1.0f - h*h) * w1[p*25 + j];
  }
  dss[i] = acc;
}

// dRt -> dR -> drij, scatter force + atom_virial
__global__ void k_bwd_scatter(const float* __restrict__ G, const float* __restrict__ dT,
                              const float* __restrict__ dss, const float* __restrict__ dstd,
                              const int* __restrict__ nbr, const float* __restrict__ rij,
                              float* __restrict__ out) {
  long i = (long)blockIdx.x * blockDim.x + threadIdx.x;
  if (i >= NPAIR) return;
  int n = (int)(i / NNEI_), k = (int)(i % NNEI_);
  int ti = n < N0A ? 0 : 1;
  float dR[4];
  for (int a = 0; a < 4; ++a) {
    float v = 0.0f;
    for (int m = 0; m < 100; ++m)
      v += G[i*100 + m] * dT[(long)n*400 + a*100 + m];
    v *= (1.0f / NNEI_);
    if (a == 0) v += dss[i];
    dR[a] = v / dstd[(long)ti*NNEI_*4 + k*4 + a];
  }
  int idx = nbr[i];
  if (idx < 0) return;
  float rv[3] = { rij[i*3+0], rij[i*3+1], rij[i*3+2] };
  float r2 = rv[0]*rv[0] + rv[1]*rv[1] + rv[2]*rv[2];
  float r  = sqrtf(r2);
  float sw = swfun(r), dsw = dswfun(r);
  float dsdr = (dsw*r - sw) / r2;          // d(sw/r)/dr
  float f    = sw / r2;                     // s/r
  float fp   = dsw / r2 - 2.0f * sw / (r2 * r);
  float g[3];
  for (int j = 0; j < 3; ++j) {
    float v = dR[0] * dsdr * rv[j] / r;
    for (int c = 0; c < 3; ++c)
      v += dR[c+1] * (fp * rv[j] * rv[c] / r + (j == c ? f : 0.0f));
    g[j] = v;
  }
  for (int j = 0; j < 3; ++j) {
    atomicAdd(&out[F_OFF + (long)idx*3 + j], -g[j]);
    atomicAdd(&out[F_OFF + (long)n  *3 + j],  g[j]);
  }
  for (int a = 0; a < 3; ++a)
    for (int b = 0; b < 3; ++b)
      atomicAdd(&out[AV_OFF + (long)idx*9 + a*3 + b], rv[a]*g[b]);
}

__global__ void k_virial_sum(float* __restrict__ out) {
  int n = blockIdx.x * blockDim.x + threadIdx.x;
  if (n >= NATOM) return;
  for (int c = 0; c < 9; ++c)
    atomicAdd(&out[V_OFF + c], out[AV_OFF + (long)n*9 + c]);
}

// ---------------- host ------------------------------------------------------
static inline unsigned blks(long n, int b = 256) { return (unsigned)((n + b - 1) / b); }

extern "C" void kernel_launch(void* const* d_in, const int* in_sizes, int n_in,
                              void* d_out, int out_size, void* d_ws, size_t ws_size,
                              hipStream_t stream) {
  const float* coord = (const float*)d_in[0];
  const float* davg  = (const float*)d_in[6];
  const float* dstd  = (const float*)d_in[7];
  const float* ew1   = (const float*)d_in[8];
  const float* eb1   = (const float*)d_in[9];
  const float* ew2   = (const float*)d_in[10];
  const float* eb2   = (const float*)d_in[11];
  const float* ew3   = (const float*)d_in[12];
  const float* eb3   = (const float*)d_in[13];
  const float* fw0   = (const float*)d_in[14];
  const float* fb0   = (const float*)d_in[15];
  const float* fw1   = (const float*)d_in[16];
  const float* fb1   = (const float*)d_in[17];
  const float* fi1   = (const float*)d_in[18];
  const float* fw2   = (const float*)d_in[19];
  const float* fb2   = (const float*)d_in[20];
  const float* fi2   = (const float*)d_in[21];
  const float* fwf   = (const float*)d_in[22];
  const float* fbf   = (const float*)d_in[23];
  const float* bae   = (const float*)d_in[24];
  float* out = (float*)d_out;

  // workspace arena
  char* w = (char*)d_ws; size_t off = 0;
  auto arena = [&](size_t nf) {
    float* p = (float*)(w + off);
    off += ((nf * sizeof(float) + 255) / 256) * 256;
    return p;
  };
  int*   nbr  = (int*)arena(NPAIR);
  float* rij  = arena(3 * NPAIR);
  float* ssb  = arena(NPAIR);
  float* Rt   = arena(4 * NPAIR);
  float* h1   = arena(25 * NPAIR);
  float* t2e  = arena(50 * NPAIR);
  float* h2e  = arena(50 * NPAIR);
  float* t3e  = arena(100 * NPAIR);
  float* G    = arena(100 * NPAIR);
  float* Tb   = arena((long)NATOM * 400);
  float* Db   = arena((long)NATOM * DWID);
  float* t0f  = arena((long)NATOM * FITW);
  float* t1f  = arena((long)NATOM * FITW);
  float* h1f  = arena((long)NATOM * FITW);
  float* t2f  = arena((long)NATOM * FITW);
  float* h2f  = arena((long)NATOM * FITW);
  float* dh2f = arena((long)NATOM * FITW);
  float* ubuf = arena((long)NATOM * FITW);
  float* dh1f = arena((long)NATOM * FITW);
  float* dh0f = arena((long)NATOM * FITW);
  float* dD   = arena((long)NATOM * DWID);
  float* dT   = arena((long)NATOM * 400);
  float* dG   = arena(100 * NPAIR);
  float* dh2e = arena(50 * NPAIR);
  float* dh1e = arena(25 * NPAIR);
  float* dss  = arena(NPAIR);

  // ---- forward ----
  k_zero<<<blks(OUT_TOTAL), 256, 0, stream>>>(out, OUT_TOTAL);
  k_build_nbr<<<NATOM, 256, 0, stream>>>(coord, nbr);
  k_build_R<<<blks(NPAIR), 256, 0, stream>>>(coord, nbr, davg, dstd, rij, Rt, ssb);
  k_emb_l1<<<blks(NPAIR), 256, 0, stream>>>(ssb, ew1, eb1, h1);

  for (int p = 0; p < 4; ++p) {
    int ti = p >> 1, tj = p & 1;
    int  a   = tj ? SEL0 : 0;
    int  seg = tj ? SEL1 : SEL0;
    long base = (long)(ti * N0A) * NNEI_ + a;
    int  M = N0A * seg;                         // multiple of 32
    dim3 g2((50 + 15) / 16, (unsigned)(M / 32));
    k_gemm_tanh_resconcat<<<g2, 32, 0, stream>>>(h1 + base*25, 25, ew2 + p*25*50, 50,
        eb2 + p*50, t2e + base*50, h2e + base*50, 50, 50, 25, seg, NNEI_);
    dim3 g3((100 + 15) / 16, (unsigned)(M / 32));
    k_gemm_tanh_resconcat<<<g3, 32, 0, stream>>>(h2e + base*50, 50, ew3 + p*50*100, 100,
        eb3 + p*100, t3e + base*100, G + base*100, 100, 100, 50, seg, NNEI_);
  }

  k_T<<<blks((long)NATOM * 400), 256, 0, stream>>>(Rt, G, Tb);
  k_D<<<blks((long)NATOM * DWID), 256, 0, stream>>>(Tb, Db);

  for (int ti = 0; ti < 2; ++ti) {
    long fb = (long)ti * N0A;
    dim3 gf(FITW / 16, N0A / 32);
    k_gemm_tanh<<<gf, 32, 0, stream>>>(Db + fb*DWID, DWID, fw0 + (long)ti*DWID*FITW, FITW,
        fb0 + ti*FITW, t0f + fb*FITW, FITW, FITW, DWID, 0, 0);
    k_gemm_fit_res<<<gf, 32, 0, stream>>>(t0f + fb*FITW, FITW, fw1 + (long)ti*FITW*FITW, FITW,
        fb1 + ti*FITW, fi1 + ti*FITW, t1f + fb*FITW, h1f + fb*FITW, FITW, FITW, FITW);
    k_gemm_fit_res<<<gf, 32, 0, stream>>>(h1f + fb*FITW, FITW, fw2 + (long)ti*FITW*FITW, FITW,
        fb2 + ti*FITW, fi2 + ti*FITW, t2f + fb*FITW, h2f + fb*FITW, FITW, FITW, FITW);
  }
  k_ae<<<blks(NATOM), 256, 0, stream>>>(h2f, fwf, fbf, bae, out);

  // ---- backward (grad wrt rij only; weights are inputs) ----
  k_dh2f_init<<<blks((long)NATOM * FITW), 256, 0, stream>>>(fwf, dh2f);
  k_fit_tanhgrad<<<blks((long)NATOM * FITW), 256, 0, stream>>>(dh2f, t2f, fi2, ubuf);
  for (int ti = 0; ti < 2; ++ti) {
    long fb = (long)ti * N0A;
    dim3 gb(FITW / 16, N0A / 32);
    k_gemm_bt<<<gb, 32, 0, stream>>>(ubuf + fb*FITW, FITW, fw2 + (long)ti*FITW*FITW, FITW,
        dh2f + fb*FITW, dh1f + fb*FITW, FITW, FITW, FITW, 0, 0);
  }
  k_fit_tanhgrad<<<blks((long)NATOM * FITW), 256, 0, stream>>>(dh1f, t1f, fi1, ubuf);
  for (int ti = 0; ti < 2; ++ti) {
    long fb = (long)ti * N0A;
    dim3 gb(FITW / 16, N0A / 32);
    k_gemm_bt<<<gb, 32, 0, stream>>>(ubuf + fb*FITW, FITW, fw1 + (long)ti*FITW*FITW, FITW,
        dh1f + fb*FITW, dh0f + fb*FITW, FITW, FITW, FITW, 0, 0);
  }
  k_fit_tanhgrad<<<blks((long)NATOM * FITW), 256, 0, stream>>>(dh0f, t0f, nullptr, ubuf);
  for (int ti = 0; ti < 2; ++ti) {
    long fb = (long)ti * N0A;
    dim3 gb(DWID / 16, N0A / 32);
    k_gemm_bt<<<gb, 32, 0, stream>>>(ubuf + fb*FITW, FITW, fw0 + (long)ti*DWID*FITW, FITW,
        nullptr, dD + fb*DWID, DWID, DWID, FITW, 0, 0);
  }

  k_dT_from_dD<<<blks((long)NATOM * 400), 256, 0, stream>>>(dD, Tb, dT);
  k_dG<<<blks(NPAIR * 100), 256, 0, stream>>>(Rt, dT, dG);

  // embedding backward: dt3 in-place into t3e, dh2e = concat-reduce(dG) + dt3@w3^T
  k_emb_tanhgrad<<<blks(NPAIR * 100), 256, 0, stream>>>(dG, t3e, t3e, NPAIR * 100);
  k_reduce_concat<<<blks(NPAIR * 50), 256, 0, stream>>>(dG, dh2e, 50, NPAIR);
  for (int p = 0; p < 4; ++p) {
    int ti = p >> 1, tj = p & 1;
    int  a   = tj ? SEL0 : 0;
    int  seg = tj ? SEL1 : SEL0;
    long base = (long)(ti * N0A) * NNEI_ + a;
    int  M = N0A * seg;
    dim3 gb((50 + 15) / 16, (unsigned)(M / 32));
    k_gemm_bt<<<gb, 32, 0, stream>>>(t3e + base*100, 100, ew3 + p*50*100, 100,
        dh2e + base*50, dh2e + base*50, 50, 50, 100, seg, NNEI_);
  }
  k_emb_tanhgrad<<<blks(NPAIR * 50), 256, 0, stream>>>(dh2e, t2e, t2e, NPAIR * 50);
  k_reduce_concat<<<blks(NPAIR * 25), 256, 0, stream>>>(dh2e, dh1e, 25, NPAIR);
  for (int p = 0; p < 4; ++p) {
    int ti = p >> 1, tj = p & 1;
    int  a   = tj ? SEL0 : 0;
    int  seg = tj ? SEL1 : SEL0;
    long base = (long)(ti * N0A) * NNEI_ + a;
    int  M = N0A * seg;
    dim3 gb((25 + 15) / 16, (unsigned)(M / 32));
    k_gemm_bt<<<gb, 32, 0, stream>>>(t2e + base*50, 50, ew2 + p*25*50, 50,
        dh1e + base*25, dh1e + base*25, 25, 25, 50, seg, NNEI_);
  }
  k_emb_l1_bwd<<<blks(NPAIR), 256, 0, stream>>>(dh1e, h1, ew1, dss);

  k_bwd_scatter<<<blks(NPAIR), 256, 0, stream>>>(G, dT, dss, dstd, nbr, rij, out);
  k_virial_sum<<<blks(NATOM), 256, 0, stream>>>(out);
}